// RNN_29042568856157
// MI455X (gfx1250) — compile-verified
//
#include <hip/hip_runtime.h>
#include <math.h>

// Problem dims (fixed by reference)
#define SEQ   16384
#define BAT   64
#define HID   20
#define VOC   29
#define MROWS (SEQ * BAT)   // 1,048,576 rows, divisible by 16

typedef float v2f __attribute__((ext_vector_type(2)));
typedef float v8f __attribute__((ext_vector_type(8)));

// The only fp32 WMMA on CDNA5: D(16x16) = A(16x4) * B(4x16) + C
__device__ __forceinline__ v8f wmma_f32(v2f a, v2f b, v8f c) {
  return __builtin_amdgcn_wmma_f32_16x16x4_f32(
      /*neg_a=*/false, a, /*neg_b=*/false, b,
      /*c_mod=*/(short)0, c, /*reuse_a=*/false, /*reuse_b=*/false);
}

__device__ __forceinline__ float fast_tanh(float x) {
#if __has_builtin(__builtin_amdgcn_tanhf)
  return __builtin_amdgcn_tanhf(x);   // v_tanh_f32 (gfx1250 transcendental)
#else
  return tanhf(x);
#endif
}

// ---------------------------------------------------------------------------
// Kernel 1: xin[m][h] = sum_v x[m][v] * W_hx[h][v] + b_hx[h] + b_hh[h]
// One wave per 16-row tile. K=29 padded to 32 (B rows >=29 are zero, so
// don't-care A lanes contribute nothing). N=20 -> two 16-wide N tiles.
// ---------------------------------------------------------------------------
__global__ __launch_bounds__(256) void
k1_input_proj(const float* __restrict__ x, const float* __restrict__ W_hx,
              const float* __restrict__ b_hx, const float* __restrict__ b_hh,
              float* __restrict__ xin) {
  const int lane = threadIdx.x & 31;
  const int hf   = lane >> 4;     // half-wave select
  const int l    = lane & 15;
  const int wid  = blockIdx.x * (blockDim.x >> 5) + (threadIdx.x >> 5);
  const long R   = (long)wid * 16;

  // B = W_hx^T, padded to 32x32. Per-lane: {B[k0][n], B[k0+1][n]}, k0=4j+2*hf.
  v2f Breg[8][2];
#pragma unroll
  for (int j = 0; j < 8; ++j) {
    const int k0 = 4 * j + 2 * hf;
#pragma unroll
    for (int nt = 0; nt < 2; ++nt) {
      const int n = nt * 16 + l;
      float b0 = 0.f, b1 = 0.f;
      if (n < HID) {
        if (k0     < VOC) b0 = W_hx[n * VOC + k0];
        if (k0 + 1 < VOC) b1 = W_hx[n * VOC + k0 + 1];
      }
      Breg[j][nt].x = b0; Breg[j][nt].y = b1;
    }
  }

  // A rows from x (row stride 29). Guard row-end so we never read OOB.
  const float* xr = x + (R + l) * (long)VOC;
  v2f Areg[8];
#pragma unroll
  for (int j = 0; j < 8; ++j) {
    const int k0 = 4 * j + 2 * hf;
    float a0 = 0.f, a1 = 0.f;
    if (k0 <= VOC - 3)      { a0 = xr[k0]; a1 = xr[k0 + 1]; } // k0 <= 26
    else if (k0 == VOC - 1) { a0 = xr[k0]; }                  // k0 == 28
    Areg[j].x = a0; Areg[j].y = a1;                           // k0 == 30: zeros
  }

  v8f c0 = {}; v8f c1 = {};
#pragma unroll
  for (int j = 0; j < 8; ++j) c0 = wmma_f32(Areg[j], Breg[j][0], c0);
#pragma unroll
  for (int j = 0; j < 8; ++j) c1 = wmma_f32(Areg[j], Breg[j][1], c1);

  const float bia0 = b_hx[l] + b_hh[l];                              // l < 16 < 20
  const float bia1 = (l < HID - 16) ? (b_hx[16 + l] + b_hh[16 + l]) : 0.f;
#pragma unroll
  for (int i = 0; i < 8; ++i) {
    const long r2 = R + i + hf * 8;          // C layout: VGPR i -> rows i / i+8
    xin[r2 * HID + l] = c0[i] + bia0;
    if (l < HID - 16) xin[r2 * HID + 16 + l] = c1[i] + bia1;
  }
}

// ---------------------------------------------------------------------------
// Kernel 2: sequential tanh recurrence. One persistent workgroup (640 thr =
// 20 waves). h double-buffered in LDS; W_hh rows in registers; xin stream
// prefetched 4 steps deep (L2-resident). Thread t -> batch b = t/10,
// hidden pair h0 = 2*(t%10).
// ---------------------------------------------------------------------------
__global__ __launch_bounds__(640, 1) void
k2_scan(const float* __restrict__ xin, const float* __restrict__ img_emb,
        const float* __restrict__ W_hh, float* __restrict__ hs) {
  __shared__ float hbuf[2][BAT * HID];   // 2 x 5120 B

  const int tid = threadIdx.x;           // 0..639
  const int b   = tid / 10;
  const int h0  = 2 * (tid % 10);

  float w0[HID], w1[HID];
#pragma unroll
  for (int k = 0; k < HID; ++k) {
    w0[k] = W_hh[h0 * HID + k];
    w1[k] = W_hh[(h0 + 1) * HID + k];
  }
  const float img0 = img_emb[b * HID + h0];
  const float img1 = img_emb[b * HID + h0 + 1];

  hbuf[0][b * HID + h0]     = 0.f;       // h_{-1} = 0
  hbuf[0][b * HID + h0 + 1] = 0.f;
  __syncthreads();

  // 4-deep register pipeline on the xin stream (8B-aligned: h0 even).
  float2 pin[4];
#pragma unroll
  for (int d = 0; d < 4; ++d)
    pin[d] = *(const float2*)&xin[((size_t)d * BAT + b) * HID + h0];

#pragma unroll 4
  for (int t = 0; t < SEQ; ++t) {
    float2 xv = pin[t & 3];
    if (t + 4 < SEQ)
      pin[t & 3] = *(const float2*)&xin[((size_t)(t + 4) * BAT + b) * HID + h0];

    const float* hc = &hbuf[t & 1][b * HID];
    float acc0 = xv.x, acc1 = xv.y;
    if (t == 1) { acc0 += img0; acc1 += img1; }   // img_emb injected at t==1
#pragma unroll
    for (int k = 0; k < HID; ++k) {
      const float hk = hc[k];
      acc0 = fmaf(hk, w0[k], acc0);               // two independent chains ->
      acc1 = fmaf(hk, w1[k], acc1);               // VOPD dual-issue candidates
    }
    const float hn0 = fast_tanh(acc0);
    const float hn1 = fast_tanh(acc1);

    float* hn = &hbuf[(t & 1) ^ 1][b * HID];
    hn[h0]     = hn0;
    hn[h0 + 1] = hn1;

    float2 st; st.x = hn0; st.y = hn1;
    *(float2*)&hs[((size_t)t * BAT + b) * HID + h0] = st;  // fire-and-forget

    __syncthreads();   // one barrier per step (double buffer)
  }
}

// ---------------------------------------------------------------------------
// Kernel 3: logits = hs @ W_oh^T + b_oh, then log_softmax over V=29.
// One wave per 16-row tile; K=20 (5 k-steps), N=29 -> two N tiles.
// Softmax reduced with width-16 shuffles (each half-wave owns one C row).
// ---------------------------------------------------------------------------
__global__ __launch_bounds__(256) void
k3_out_softmax(const float* __restrict__ hs, const float* __restrict__ W_oh,
               const float* __restrict__ b_oh, float* __restrict__ out) {
  const int lane = threadIdx.x & 31;
  const int hf   = lane >> 4;
  const int l    = lane & 15;
  const int wid  = blockIdx.x * (blockDim.x >> 5) + (threadIdx.x >> 5);
  const long R   = (long)wid * 16;

  // B = W_oh^T: B[k=h][n=v] = W_oh[v*20 + h]; k always < 20.
  v2f Breg[5][2];
#pragma unroll
  for (int j = 0; j < 5; ++j) {
    const int k0 = 4 * j + 2 * hf;     // <= 18
#pragma unroll
    for (int nt = 0; nt < 2; ++nt) {
      const int n = nt * 16 + l;
      float b0 = 0.f, b1 = 0.f;
      if (n < VOC) { b0 = W_oh[n * HID + k0]; b1 = W_oh[n * HID + k0 + 1]; }
      Breg[j][nt].x = b0; Breg[j][nt].y = b1;
    }
  }

  const float* hr = hs + (R + l) * (long)HID;
  v2f Areg[5];
#pragma unroll
  for (int j = 0; j < 5; ++j) {
    const int k0 = 4 * j + 2 * hf;     // k0+1 <= 19: always in-row
    Areg[j].x = hr[k0]; Areg[j].y = hr[k0 + 1];
  }

  v8f c0 = {}; v8f c1 = {};
#pragma unroll
  for (int j = 0; j < 5; ++j) c0 = wmma_f32(Areg[j], Breg[j][0], c0);
#pragma unroll
  for (int j = 0; j < 5; ++j) c1 = wmma_f32(Areg[j], Breg[j][1], c1);

  const float bo0 = b_oh[l];                                  // l < 16 < 29
  const float bo1 = (l < VOC - 16) ? b_oh[16 + l] : 0.f;      // VOC-16 = 13
#pragma unroll
  for (int i = 0; i < 8; ++i) {
    float v0 = c0[i] + bo0;
    float v1 = (l < VOC - 16) ? (c1[i] + bo1) : -INFINITY;
    // max over this half-wave's 16 lanes (row i for lanes<16, row i+8 above)
    float m = fmaxf(v0, v1);
#pragma unroll
    for (int d = 1; d < 16; d <<= 1) m = fmaxf(m, __shfl_xor(m, d, 16));
    float e = __expf(v0 - m) + ((l < VOC - 16) ? __expf(v1 - m) : 0.f);
#pragma unroll
    for (int d = 1; d < 16; d <<= 1) e += __shfl_xor(e, d, 16);
    const float lse = m + __logf(e);

    const long r2 = R + i + hf * 8;
    out[r2 * VOC + l] = v0 - lse;
    if (l < VOC - 16) out[r2 * VOC + 16 + l] = v1 - lse;
  }
}

// ---------------------------------------------------------------------------
extern "C" void kernel_launch(void* const* d_in, const int* in_sizes, int n_in,
                              void* d_out, int out_size, void* d_ws, size_t ws_size,
                              hipStream_t stream) {
  (void)in_sizes; (void)n_in; (void)out_size; (void)ws_size;

  const float* x       = (const float*)d_in[0];  // [S,B,V]
  const float* img_emb = (const float*)d_in[1];  // [B,H]
  const float* W_hx    = (const float*)d_in[2];  // [1,H,V]
  const float* b_hx    = (const float*)d_in[3];  // [1,H]
  const float* W_hh    = (const float*)d_in[4];  // [1,H,H]
  const float* b_hh    = (const float*)d_in[5];  // [1,H]
  const float* W_oh    = (const float*)d_in[6];  // [V,H]
  const float* b_oh    = (const float*)d_in[7];  // [V]

  float* out = (float*)d_out;
  // xin (M*20 floats = 84 MB) lives in d_out (121.6 MB); fully overwritten by k3.
  float* xin = out;
  // hs (M*20 floats = 84 MB) lives in the workspace.
  float* hs  = (float*)d_ws;

  const int tiles          = MROWS / 16;           // 65536 wave-tiles
  const int waves_per_blk  = 8;                    // 256 threads
  const int blocks         = tiles / waves_per_blk;

  k1_input_proj<<<blocks, 256, 0, stream>>>(x, W_hx, b_hx, b_hh, xin);
  k2_scan<<<1, 640, 0, stream>>>(xin, img_emb, W_hh, hs);
  k3_out_softmax<<<blocks, 256, 0, stream>>>(hs, W_oh, b_oh, out);
}